// GCNNet_86483461472760
// MI455X (gfx1250) — compile-verified
//
#include <hip/hip_runtime.h>

#define NN 20000
#define NE 320000
#define INC 128
#define HIDC 256
#define OUTC 32
#define SLOPE 0.01f

typedef __attribute__((ext_vector_type(2))) float v2f;
typedef __attribute__((ext_vector_type(8))) float v8f;
typedef int v4i __attribute__((vector_size(16)));   // matches builtin's pointee type

#ifndef __has_builtin
#define __has_builtin(x) 0
#endif
#if __has_builtin(__builtin_amdgcn_global_load_async_to_lds_b128) && \
    __has_builtin(__builtin_amdgcn_s_wait_asynccnt)
#define HAVE_ASYNC 1
#else
#define HAVE_ASYNC 0
#endif

// global 16B -> LDS 16B; async DMA path (ASYNCcnt-tracked) on CDNA5
__device__ __forceinline__ void cp_g2l(float* ldsDst, const float* gSrc) {
#if HAVE_ASYNC
    __builtin_amdgcn_global_load_async_to_lds_b128((v4i*)gSrc, (v4i*)ldsDst, 0, 0);
#else
    *(float4*)ldsDst = *(const float4*)gSrc;
#endif
}

// ---------------- degree / norm ----------------
__global__ void deg_kernel(const int* __restrict__ dst, int* __restrict__ deg, int nE) {
    int i = blockIdx.x * blockDim.x + threadIdx.x;
    if (i < nE) atomicAdd(&deg[dst[i]], 1);
}

__global__ void dinv_kernel(const int* __restrict__ deg, float* __restrict__ dinv, int n) {
    int i = blockIdx.x * blockDim.x + threadIdx.x;
    if (i < n) {
        int d = deg[i];
        dinv[i] = (d > 0) ? rsqrtf((float)d) : 0.0f;
    }
}

__global__ void norm_kernel(const int* __restrict__ src, const int* __restrict__ dst,
                            const float* __restrict__ dinv, float* __restrict__ norm, int nE) {
    int i = blockIdx.x * blockDim.x + threadIdx.x;
    if (i < nE) norm[i] = dinv[src[i]] * dinv[dst[i]];
}

// ---------------- LDS double-buffered WMMA f32 GEMM ----------------
// C[M,N] = A[M,K] @ W[K,N].  Block = 320 threads = 10 waves; block tile 160 x (16*NTILES).
// M = 20000 = 125*160 and N in {256,32} tile exactly -> EXEC stays all-ones (WMMA requirement).
template <int NTILES>
__global__ __launch_bounds__(320) void gemm_wmma_lds(const float* __restrict__ A,
                                                     const float* __restrict__ W,
                                                     float* __restrict__ C,
                                                     int M, int K, int N) {
    constexpr int BLKM = 160;
    constexpr int KT   = 16;              // K-slice per stage
    constexpr int BLKN = 16 * NTILES;
    constexpr int SA   = KT + 4;          // 20 floats: 16B-aligned rows, conflict-free
    constexpr int SB   = BLKN + 8;        // 72/40 floats: halves land on disjoint banks
    constexpr int ASZ  = BLKM * SA;
    constexpr int BF4  = KT * BLKN / 4;   // 256 / 128 (power of two)
    constexpr int BROW4 = BLKN / 4;
    constexpr int BSZ  = KT * SB;

    __shared__ float lds[2][ASZ + BSZ];

    const int tid    = threadIdx.x;
    const int lane   = tid & 31;
    const int wave   = tid >> 5;
    const int laneLo = lane & 15;
    const int laneHi = lane >> 4;
    const int nStrips = N / BLKN;
    const int mBase  = (blockIdx.x / nStrips) * BLKM;
    const int nBase  = (blockIdx.x % nStrips) * BLKN;
    const int nStages = K / KT;
    (void)M;

    // per-thread copy assignments: 2 A-chunks + 1 B-chunk (uniform -> fixed asynccnt)
    const int ac0 = tid, ac1 = tid + 320;
    const int bc  = tid & (BF4 - 1);

    auto loadStage = [&](int s, int buf) {
        {
            int row = ac0 >> 2, c4 = ac0 & 3;
            cp_g2l(&lds[buf][row * SA + c4 * 4],
                   A + (size_t)(mBase + row) * K + s * KT + c4 * 4);
        }
        {
            int row = ac1 >> 2, c4 = ac1 & 3;
            cp_g2l(&lds[buf][row * SA + c4 * 4],
                   A + (size_t)(mBase + row) * K + s * KT + c4 * 4);
        }
        {
            int row = bc / BROW4, c4 = bc % BROW4;
            cp_g2l(&lds[buf][ASZ + row * SB + c4 * 4],
                   W + (size_t)(s * KT + row) * N + nBase + c4 * 4);
        }
    };

    v8f acc[NTILES] = {};

    loadStage(0, 0);
    for (int s = 0; s < nStages; ++s) {
        const int nxt = s + 1;
        if (nxt < nStages) loadStage(nxt, nxt & 1);   // prefetch next slice
#if HAVE_ASYNC
        if (nxt < nStages) __builtin_amdgcn_s_wait_asynccnt(3);  // next stage stays in flight
        else               __builtin_amdgcn_s_wait_asynccnt(0);
#endif
        __syncthreads();

        const float* As = lds[s & 1];
        const float* Bs = lds[s & 1] + ASZ;
        const int arow = wave * 16 + laneLo;
#pragma unroll
        for (int kk = 0; kk < KT; kk += 4) {
            const int ka = kk + 2 * laneHi;
            v2f a;
            a.x = As[arow * SA + ka];
            a.y = As[arow * SA + ka + 1];
#pragma unroll
            for (int t = 0; t < NTILES; ++t) {
                const int col = t * 16 + laneLo;
                v2f b;
                b.x = Bs[ka * SB + col];
                b.y = Bs[(ka + 1) * SB + col];
                acc[t] = __builtin_amdgcn_wmma_f32_16x16x4_f32(
                    false, a, false, b, (short)0, acc[t], false, false);
            }
        }
        __syncthreads();
    }

    const int rBase = mBase + wave * 16 + 8 * laneHi;
#pragma unroll
    for (int t = 0; t < NTILES; ++t) {
#pragma unroll
        for (int r = 0; r < 8; ++r) {
            C[(size_t)(rBase + r) * N + nBase + t * 16 + laneLo] = acc[t][r];
        }
    }
}

// ---------------- edge scatter: out[dst] += h[src] * norm ----------------
__global__ void scatter_kernel(const float* __restrict__ h, const int* __restrict__ src,
                               const int* __restrict__ dst, const float* __restrict__ norm,
                               float* __restrict__ out, int nE, int C) {
    int tid = blockIdx.x * blockDim.x + threadIdx.x;
    int vecPerEdge = C >> 2;
    int e = tid / vecPerEdge;
    int v = tid - e * vecPerEdge;
    if (e >= nE) return;
    int s = src[e], d = dst[e];
    float nw = norm[e];
    float4 m = *((const float4*)(h + (size_t)s * C) + v);
    float* op = out + (size_t)d * C + v * 4;
    unsafeAtomicAdd(op + 0, m.x * nw);
    unsafeAtomicAdd(op + 1, m.y * nw);
    unsafeAtomicAdd(op + 2, m.z * nw);
    unsafeAtomicAdd(op + 3, m.w * nw);
}

// ---------------- bias + leaky relu (in place) ----------------
__global__ void bias_leaky_kernel(float* __restrict__ h, const float* __restrict__ b,
                                  int total, int Cmask) {
    int i = blockIdx.x * blockDim.x + threadIdx.x;
    if (i < total) {
        float v = h[i] + b[i & Cmask];
        h[i] = (v > 0.0f) ? v : SLOPE * v;
    }
}

// ---------------- first node per graph ----------------
__global__ void firstidx_kernel(const int* __restrict__ batch, int* __restrict__ fidx, int n) {
    int i = blockIdx.x * blockDim.x + threadIdx.x;
    if (i < n) atomicMin(&fidx[batch[i]], i);
}

// ---------------- final gather + b3 ----------------
__global__ void gather_kernel(const float* __restrict__ h, const int* __restrict__ fidx,
                              const float* __restrict__ b, float* __restrict__ out,
                              int nG, int C) {
    int i = blockIdx.x * blockDim.x + threadIdx.x;
    if (i < nG * C) {
        int g = i / C, c = i - g * C;
        out[i] = h[(size_t)fidx[g] * C + c] + b[c];
    }
}

static inline size_t alignUp(size_t x) { return (x + 255) & ~(size_t)255; }

extern "C" void kernel_launch(void* const* d_in, const int* in_sizes, int n_in,
                              void* d_out, int out_size, void* d_ws, size_t ws_size,
                              hipStream_t stream) {
    const float* x    = (const float*)d_in[0];
    const int*   src  = (const int*)d_in[1];
    const int*   dst  = (const int*)d_in[2];
    const int*   batch= (const int*)d_in[3];
    const float* W1   = (const float*)d_in[4];
    const float* b1   = (const float*)d_in[5];
    const float* W2   = (const float*)d_in[6];
    const float* b2   = (const float*)d_in[7];
    const float* W3   = (const float*)d_in[8];
    const float* b3   = (const float*)d_in[9];
    const int nG = out_size / OUTC;   // 100
    float* out = (float*)d_out;

    // ---- workspace carve-out ----
    char* ws = (char*)d_ws;
    size_t off = 0;
    int*   deg  = (int*)(ws + off);   off += alignUp((size_t)NN * 4);
    float* dinv = (float*)(ws + off); off += alignUp((size_t)NN * 4);
    float* norm = (float*)(ws + off); off += alignUp((size_t)NE * 4);
    int*   fidx = (int*)(ws + off);   off += alignUp(4096);
    float* bufA = (float*)(ws + off); off += alignUp((size_t)NN * HIDC * 4);
    float* bufB = (float*)(ws + off); off += alignUp((size_t)NN * HIDC * 4);
    float* bufC = (float*)(ws + off); off += alignUp((size_t)NN * OUTC * 4);
    (void)ws_size; (void)n_in; (void)in_sizes;

    const int B = 256;

    // degree / normalization (shared by all layers)
    (void)hipMemsetAsync(deg, 0, (size_t)NN * 4, stream);
    deg_kernel<<<(NE + B - 1) / B, B, 0, stream>>>(dst, deg, NE);
    dinv_kernel<<<(NN + B - 1) / B, B, 0, stream>>>(deg, dinv, NN);
    norm_kernel<<<(NE + B - 1) / B, B, 0, stream>>>(src, dst, dinv, norm, NE);

    // first node per graph
    (void)hipMemsetAsync(fidx, 0x7f, 4096, stream);  // 0x7f7f7f7f > NN
    firstidx_kernel<<<(NN + B - 1) / B, B, 0, stream>>>(batch, fidx, NN);

    // ---- layer 1: h = leaky(scatter(x @ W1) + b1) ----
    gemm_wmma_lds<4><<<(NN / 160) * (HIDC / 64), 320, 0, stream>>>(x, W1, bufA, NN, INC, HIDC);
    (void)hipMemsetAsync(bufB, 0, (size_t)NN * HIDC * 4, stream);
    scatter_kernel<<<((size_t)NE * (HIDC / 4) + B - 1) / B, B, 0, stream>>>(bufA, src, dst, norm, bufB, NE, HIDC);
    bias_leaky_kernel<<<((size_t)NN * HIDC + B - 1) / B, B, 0, stream>>>(bufB, b1, NN * HIDC, HIDC - 1);

    // ---- layer 2 ----
    gemm_wmma_lds<4><<<(NN / 160) * (HIDC / 64), 320, 0, stream>>>(bufB, W2, bufA, NN, HIDC, HIDC);
    (void)hipMemsetAsync(bufB, 0, (size_t)NN * HIDC * 4, stream);   // stream-ordered after gemm
    scatter_kernel<<<((size_t)NE * (HIDC / 4) + B - 1) / B, B, 0, stream>>>(bufA, src, dst, norm, bufB, NE, HIDC);
    bias_leaky_kernel<<<((size_t)NN * HIDC + B - 1) / B, B, 0, stream>>>(bufB, b2, NN * HIDC, HIDC - 1);

    // ---- layer 3 (no activation; b3 fused into final gather) ----
    gemm_wmma_lds<2><<<(NN / 160) * (OUTC / 32), 320, 0, stream>>>(bufB, W3, bufA, NN, HIDC, OUTC);
    (void)hipMemsetAsync(bufC, 0, (size_t)NN * OUTC * 4, stream);
    scatter_kernel<<<((size_t)NE * (OUTC / 4) + B - 1) / B, B, 0, stream>>>(bufA, src, dst, norm, bufC, NE, OUTC);

    gather_kernel<<<(nG * OUTC + B - 1) / B, B, 0, stream>>>(bufC, fidx, b3, out, nG, OUTC);
}